// SampledNeighbor_52596169507059
// MI455X (gfx1250) — compile-verified
//
#include <hip/hip_runtime.h>
#include <hip/hip_bf16.h>

// Problem constants (from reference)
#define NTOK   100000
#define NH     256
#define BSZ    1024
#define NSAMP  32

// Tiling
#define TILE_M 16          // rows per block (one WMMA M tile)
#define WAVES  8           // 256 threads / wave32
#define NSLICE 8           // i-dimension split across blockIdx.y
#define IPB    (BSZ / NSLICE)       // 128 i's per block
#define GROUPS (IPB / WAVES)        // 16 sample-groups per wave
#define CHUNK  16                   // H columns staged per chunk
#define NCHUNK (NH / CHUNK)         // 16
#define KSTEPS (CHUNK / 4)          // 4 WMMA K-steps per chunk (K=4 each)
#define XSTRIDE 260        // 256 + 4 DWORD TDM pad (bank-spread, 16B rows)
#define WSTRIDE 20         // 16 + 4 DWORD TDM pad (bank-spread, 16B rows)
#define SLAB    (NSAMP * WSTRIDE)   // one chunk buffer: 32 rows x 20 f32

typedef float v2f __attribute__((ext_vector_type(2)));
typedef float v8f __attribute__((ext_vector_type(8)));
typedef unsigned int u32x4 __attribute__((ext_vector_type(4)));
typedef int i32x4 __attribute__((ext_vector_type(4)));
typedef int i32x8 __attribute__((ext_vector_type(8)));

// ---------------------------------------------------------------------------
// TDM descriptor helpers (CDNA5 ISA ch.8).  amdgpu-toolchain 6-arg builtin.
// ---------------------------------------------------------------------------

// Gather 8 rows (32-bit indices) of CHUNK f32 from the [NTOK, NH] f32 weight
// tensor into LDS at lds_byte; hardware-pads each 16-DWORD row by 4 DWORDs
// (stride 20 == WSTRIDE, conflict-free for 16-lane B-fragment reads).
__device__ __forceinline__ void tdm_gather8(unsigned lds_byte,
                                            unsigned long long gaddr,
                                            i32x4 idx_lo, i32x4 idx_hi) {
    u32x4 g0;
    g0.x = 0xC0000001u;                          // count=1, gather, 32b idx
    g0.y = lds_byte;                             // lds_addr
    g0.z = (unsigned)gaddr;                      // global_addr[31:0]
    g0.w = (unsigned)(gaddr >> 32) | 0x80000000u;// global_addr[56:32], type=2
    i32x8 g1;
    g1[0] = 0x06D20000;          // data_size=4B, pad_en, intv=16dw, amt=4dw
    g1[1] = (NH & 0xFFFF) << 16;                 // tensor_dim0 = NH
    g1[2] = (int)(((NTOK & 0xFFFF) << 16) | (NH >> 16));  // tensor_dim1 lo
    g1[3] = (CHUNK << 16) | (NTOK >> 16);        // tile_dim0=16, dim1 hi
    g1[4] = 8;                                   // tile_dim1 = #indices
    g1[5] = NH;                                  // tensor_dim0_stride
    g1[6] = 0;
    g1[7] = 0;
    i32x8 z8 = {0, 0, 0, 0, 0, 0, 0, 0};
    __builtin_amdgcn_tensor_load_to_lds(g0, g1, idx_lo, idx_hi, z8, 0);
}

// Plain 2D tile load: TILE_M rows x NH f32, padded 256->260 DWORD rows.
__device__ __forceinline__ void tdm_load_xtile(unsigned lds_byte,
                                               unsigned long long gaddr) {
    u32x4 g0;
    g0.x = 1u;                                   // count=1
    g0.y = lds_byte;
    g0.z = (unsigned)gaddr;
    g0.w = (unsigned)(gaddr >> 32) | 0x80000000u;
    i32x8 g1;
    g1[0] = 0x07D20000;          // data_size=4B, pad_en, intv=256dw, amt=4dw
    g1[1] = (NH & 0xFFFF) << 16;                 // tensor_dim0 = NH
    g1[2] = (BSZ & 0xFFFF) << 16;                // tensor_dim1 = BSZ
    g1[3] = (NH << 16);                          // tile_dim0 = NH
    g1[4] = TILE_M;                              // tile_dim1 = 16
    g1[5] = NH;                                  // tensor_dim0_stride
    g1[6] = 0;
    g1[7] = 0;
    i32x4 z = {0, 0, 0, 0};
    i32x8 z8 = {0, 0, 0, 0, 0, 0, 0, 0};
    __builtin_amdgcn_tensor_load_to_lds(g0, g1, z, z, z8, 0);
}

// ---------------------------------------------------------------------------
// Kernel 1: wsum[h] = sum_i weight[labels[i]][h]
// ---------------------------------------------------------------------------
__global__ void sn_wsum_kernel(const float* __restrict__ weight,
                               const int* __restrict__ labels,
                               float* __restrict__ wsum) {
    const int h = threadIdx.x;          // 256 threads, one per column
    float s = 0.0f;
    for (int i = 0; i < BSZ; ++i) {
        const int row = labels[i];
        s += weight[(size_t)row * NH + h];   // coalesced across threads
    }
    wsum[h] = s;
}

// ---------------------------------------------------------------------------
// Kernel 2: main sampled-logsumexp GEMM (double-buffered TDM-fed WMMA).
// partial[slice*BSZ + b] = sum_{i in slice} log( sum_k exp(x_b . w_{sid[i,k]}) )
// ---------------------------------------------------------------------------
__global__ __launch_bounds__(256, 1)
void sn_main_kernel(const float* __restrict__ x,
                    const int* __restrict__ sids,
                    const float* __restrict__ weight,
                    float* __restrict__ partial) {
    __shared__ float xt[TILE_M * XSTRIDE];             // 16.6 KB (TDM-written)
    __shared__ float wt[WAVES * 2 * SLAB];             // 40.0 KB (TDM-written)
    __shared__ float red[WAVES][TILE_M];               // 0.5 KB

    const int rowTile = blockIdx.x;     // 0..63
    const int slice   = blockIdx.y;     // 0..7
    const int t       = threadIdx.x;
    const int wave    = t >> 5;
    const int lane    = t & 31;
    const int rowBase = rowTile * TILE_M;

    // ---- X tile via one TDM 2D descriptor (wave 0 issues, all consume) ----
    if (wave == 0) {
        const unsigned long long xg =
            (unsigned long long)(uintptr_t)(x + (size_t)rowBase * NH);
        tdm_load_xtile((unsigned)(uintptr_t)(void*)xt, xg);
        __builtin_amdgcn_s_wait_tensorcnt(0);
    }
    __syncthreads();
    asm volatile("" ::: "memory");   // TDM wrote LDS behind the compiler

    // two ping-pong chunk buffers per wave (pointers stay addrspace(3))
    float* wb0 = wt + wave * (2 * SLAB);
    float* wb1 = wb0 + SLAB;
    const unsigned wlds0 = (unsigned)(uintptr_t)(void*)wb0;
    const unsigned wlds1 = (unsigned)(uintptr_t)(void*)wb1;

    const int m    = lane & 15;               // M (A) / N (B,C,D) index
    const int half = lane >> 4;               // lane half selects K pair

    // ---- hoist all A fragments for the full K=256 into registers ----
    v2f afr[NCHUNK * KSTEPS];                 // 64 x v2f = 128 VGPRs
    #pragma unroll
    for (int kq = 0; kq < NCHUNK * KSTEPS; ++kq)
        afr[kq] = *(const v2f*)(xt + m * XSTRIDE + kq * 4 + 2 * half);

    float logacc[8];
    #pragma unroll
    for (int v = 0; v < 8; ++v) logacc[v] = 0.0f;

    const unsigned long long wbase = (unsigned long long)(uintptr_t)weight;

    for (int g = 0; g < GROUPS; ++g) {
        const int i = slice * IPB + g * WAVES + wave;   // sample-group id

        // one coalesced load + readlane -> wave-uniform ids in SGPRs
        const int myid = sids[i * NSAMP + lane];
        int sidu[NSAMP];
        #pragma unroll
        for (int j = 0; j < NSAMP; ++j)
            sidu[j] = __builtin_amdgcn_readlane(myid, j);

        v8f acc0 = {};   // columns 0..15 of this group
        v8f acc1 = {};   // columns 16..31

        // prologue: start chunk 0 into buffer 0
        #pragma unroll
        for (int d = 0; d < 4; ++d) {
            i32x4 ilo = {sidu[d*8+0], sidu[d*8+1], sidu[d*8+2], sidu[d*8+3]};
            i32x4 ihi = {sidu[d*8+4], sidu[d*8+5], sidu[d*8+6], sidu[d*8+7]};
            tdm_gather8(wlds0 + d * 8 * WSTRIDE * 4, wbase, ilo, ihi);
        }

        for (int ch = 0; ch < NCHUNK; ++ch) {
            const int cur = ch & 1;
            // issue next chunk's gathers into the other buffer
            if (ch + 1 < NCHUNK) {
                const unsigned nlds = (cur ? wlds0 : wlds1);
                const unsigned long long ga =
                    wbase + (unsigned)((ch + 1) * CHUNK * 4);
                #pragma unroll
                for (int d = 0; d < 4; ++d) {
                    i32x4 ilo = {sidu[d*8+0], sidu[d*8+1],
                                 sidu[d*8+2], sidu[d*8+3]};
                    i32x4 ihi = {sidu[d*8+4], sidu[d*8+5],
                                 sidu[d*8+6], sidu[d*8+7]};
                    tdm_gather8(nlds + d * 8 * WSTRIDE * 4, ga, ilo, ihi);
                }
                // in-order TENSORcnt: <=4 means current chunk's 4 are done
                __builtin_amdgcn_s_wait_tensorcnt(4);
            } else {
                __builtin_amdgcn_s_wait_tensorcnt(0);
            }
            asm volatile("" ::: "memory");

            const float* wcur = cur ? wb1 : wb0;
            #pragma unroll
            for (int ks = 0; ks < KSTEPS; ++ks) {
                const int kk = ks * 4 + 2 * half;   // K pair per lane half
                // B frag: lane(n,half) holds w_{sid[n]}[k0+2h .. +1]
                v2f b0 = *(const v2f*)(wcur + m * WSTRIDE + kk);
                v2f b1 = *(const v2f*)(wcur + (16 + m) * WSTRIDE + kk);
                const v2f a = afr[ch * KSTEPS + ks];
                acc0 = __builtin_amdgcn_wmma_f32_16x16x4_f32(
                           false, a, false, b0, (short)0, acc0, false, false);
                acc1 = __builtin_amdgcn_wmma_f32_16x16x4_f32(
                           false, a, false, b1, (short)0, acc1, false, false);
            }
            __builtin_amdgcn_wave_barrier();
        }

        // exp + 32-column segment sum. C/D layout: lane = N, VGPR v -> row
        // m = v + 8*half. Each lane sums its 2 columns, then 16-lane xor tree.
        #pragma unroll
        for (int v = 0; v < 8; ++v) {
            float e = __expf(acc0[v]) + __expf(acc1[v]);
            e += __shfl_xor(e, 8, 32);
            e += __shfl_xor(e, 4, 32);
            e += __shfl_xor(e, 2, 32);
            e += __shfl_xor(e, 1, 32);
            if (m == 0) logacc[v] += __logf(e);
        }
    }

    // lanes 0 and 16 hold per-row partial logsums (rows v and v+8)
    if (m == 0) {
        #pragma unroll
        for (int v = 0; v < 8; ++v) red[wave][v + 8 * half] = logacc[v];
    }
    __syncthreads();

    if (t < TILE_M) {
        float s = 0.0f;
        #pragma unroll
        for (int w = 0; w < WAVES; ++w) s += red[w][t];
        partial[slice * BSZ + rowBase + t] = s;
    }
}

// ---------------------------------------------------------------------------
// Kernel 3: out[b] = sum_slices partial[s][b] - x_b . wsum
// ---------------------------------------------------------------------------
__global__ void sn_final_kernel(const float* __restrict__ x,
                                const float* __restrict__ wsum,
                                const float* __restrict__ partial,
                                float* __restrict__ out) {
    __shared__ float ws[NH];
    const int b = blockIdx.x * blockDim.x + threadIdx.x;
    for (int h = threadIdx.x; h < NH; h += blockDim.x) ws[h] = wsum[h];
    __syncthreads();

    const float* __restrict__ xr = x + (size_t)b * NH;
    float dot = 0.0f;
    #pragma unroll 4
    for (int h = 0; h < NH; h += 4) {
        const float4 xv = *(const float4*)(xr + h);
        dot += xv.x * ws[h] + xv.y * ws[h + 1] +
               xv.z * ws[h + 2] + xv.w * ws[h + 3];
    }
    float s = 0.0f;
    #pragma unroll
    for (int sl = 0; sl < NSLICE; ++sl) s += partial[sl * BSZ + b];
    out[b] = s - dot;
}

// ---------------------------------------------------------------------------
extern "C" void kernel_launch(void* const* d_in, const int* in_sizes, int n_in,
                              void* d_out, int out_size, void* d_ws, size_t ws_size,
                              hipStream_t stream) {
    const float* x      = (const float*)d_in[0];   // inputs  [B, NH]
    const int*   labels = (const int*)  d_in[1];   // labels  [B]
    const int*   sids   = (const int*)  d_in[2];   // sample_ids [B, K]
    const float* weight = (const float*)d_in[3];   // weight  [NTOK, NH]
    float*       out    = (float*)d_out;           // [B]

    float* wsum    = (float*)d_ws;                 // NH floats
    float* partial = wsum + NH;                    // NSLICE * BSZ floats

    sn_wsum_kernel<<<1, 256, 0, stream>>>(weight, labels, wsum);

    dim3 grid(BSZ / TILE_M, NSLICE);               // 64 x 8 = 512 blocks
    sn_main_kernel<<<grid, 256, 0, stream>>>(x, sids, weight, partial);

    sn_final_kernel<<<BSZ / 256, 256, 0, stream>>>(x, wsum, partial, out);
}